// LocalRmsNorm_89343909692118
// MI455X (gfx1250) — compile-verified
//
#include <hip/hip_runtime.h>
#include <cmath>

typedef float v2f __attribute__((ext_vector_type(2)));
typedef float v4f __attribute__((ext_vector_type(4)));
typedef float v8f __attribute__((ext_vector_type(8)));

#define HH   128
#define WW   128
#define DD   256
#define CCH  16      // channels per workgroup
#define KW   7
#define RAD  3
#define HSEG 32      // output rows per workgroup
#define EPSV 1e-7f

__global__ __launch_bounds__(256) void LocalRmsNorm_kernel(
    const float* __restrict__ x, const float* __restrict__ weight,
    float* __restrict__ out)
{
    __shared__ float vs[WW * CCH];   // vertical 7-sums for one row: [w][c], 8 KiB

    const int tid  = threadIdx.x;
    const int bx   = blockIdx.x;
    const int hseg = bx & 3;
    const int cch  = (bx >> 2) & 15;
    const int b    = bx >> 6;
    const int cbase = cch * CCH;
    const int h0    = hseg * HSEG;

    // ---------------- producer mapping: (w, 4 ch) x 2 w-positions ----------
    const int c4 = (tid & 3) * 4;          // channel offset within chunk
    const int wq = tid >> 2;               // 0..63 -> handles w=wq and w=wq+64
    const size_t xb   = (size_t)b * (HH * WW * DD);
    const size_t colA = xb + (size_t)wq * DD + cbase + c4;
    const size_t colB = xb + (size_t)(wq + 64) * DD + cbase + c4;

    // ---------------- consumer (WMMA) per-wave constants --------------------
    const int lane  = tid & 31;
    const int wave  = tid >> 5;            // 0..7 -> w-tile
    const int w0    = wave * 16;
    const int n     = lane & 15;           // N (channel in chunk) == M for A
    const int khalf = lane >> 4;

    // Band matrix A tiles (6 x 16x4) with w-boundary zeroing folded in,
    // plus clamped LDS byte offsets for the B fragments.
    v2f Aq[6];
    int Boff[6][2];
    #pragma unroll
    for (int t = 0; t < 6; ++t) {
        #pragma unroll
        for (int kk = 0; kk < 2; ++kk) {
            int p    = 4 * t + 2 * khalf + kk;   // K index in 24-wide window
            int w_in = w0 - 4 + p;               // input w position
            bool band = (p >= n + 1) && (p <= n + 7);   // |w_out - w_in| <= 3
            bool inb  = (w_in >= 0) && (w_in < WW);
            float a = (band && inb) ? 1.0f : 0.0f;
            if (kk == 0) Aq[t].x = a; else Aq[t].y = a;
            int wc = w_in < 0 ? 0 : (w_in > WW - 1 ? WW - 1 : w_in);
            Boff[t][kk] = (wc * CCH + n) * 4;
        }
    }
    const float wgt = weight[cbase + n];

    // ---------------- vertical sliding window state -------------------------
    v4f winA[KW], winB[KW];
    #pragma unroll
    for (int i = 0; i < KW; ++i) { v4f z = {}; winA[i] = z; winB[i] = z; }
    #pragma unroll
    for (int i = 0; i < KW - 1; ++i) {           // rows h0-3 .. h0+2
        int hh = h0 - RAD + i;
        if (hh >= 0 && hh < HH) {
            v4f xa = *(const v4f*)(x + colA + (size_t)hh * (WW * DD));
            v4f xc = *(const v4f*)(x + colB + (size_t)hh * (WW * DD));
            winA[i] = xa * xa;
            winB[i] = xc * xc;
        }
    }

    for (int h = h0; h < h0 + HSEG; ++h) {
        // load row h+3 (zero pad past bottom)
        {
            int hh = h + RAD;
            v4f xa = {}, xc = {};
            if (hh < HH) {
                xa = *(const v4f*)(x + colA + (size_t)hh * (WW * DD));
                xc = *(const v4f*)(x + colB + (size_t)hh * (WW * DD));
            }
            winA[KW - 1] = xa * xa;
            winB[KW - 1] = xc * xc;
            // stream-ahead prefetch (gfx1250 global_prefetch_b8)
            int hp = (hh + 4 < HH) ? hh + 4 : HH - 1;
            __builtin_prefetch(x + colA + (size_t)hp * (WW * DD), 0, 1);
        }
        // vertical 7-sum
        v4f sA = winA[0], sB = winB[0];
        #pragma unroll
        for (int i = 1; i < KW; ++i) { sA += winA[i]; sB += winB[i]; }
        *(v4f*)(vs + wq * CCH + c4)        = sA;
        *(v4f*)(vs + (wq + 64) * CCH + c4) = sB;
        __syncthreads();

        // horizontal 7-sum: banded matmul, 6 chained f32 WMMAs
        v8f acc = {};
        const char* vsb = (const char*)vs;
        #pragma unroll
        for (int t = 0; t < 6; ++t) {
            v2f bf;
            bf.x = *(const float*)(vsb + Boff[t][0]);
            bf.y = *(const float*)(vsb + Boff[t][1]);
            acc = __builtin_amdgcn_wmma_f32_16x16x4_f32(
                      false, Aq[t], false, bf, (short)0, acc, false, false);
        }

        // normalize + store: D vgpr r -> w_out = w0 + r + 8*khalf, channel n
        // rms^-1 via v_rsq_f32 (TRANS op, ~1 ulp; co-executes with VALU)
        #pragma unroll
        for (int r = 0; r < 8; ++r) {
            int w_out  = w0 + r + 8 * khalf;
            size_t idx = xb + ((size_t)h * WW + w_out) * DD + cbase + n;
            float xv   = x[idx];                    // L2-hot reload
            float m    = fmaf(acc[r], 1.0f / 49.0f, EPSV);
            float rinv = __builtin_amdgcn_rsqf(m);  // v_rsq_f32
            out[idx]   = xv * wgt * rinv;
        }
        __syncthreads();   // protect vs before next row's producer writes

        // rotate the vertical window
        #pragma unroll
        for (int i = 0; i < KW - 1; ++i) { winA[i] = winA[i + 1]; winB[i] = winB[i + 1]; }
    }
}

extern "C" void kernel_launch(void* const* d_in, const int* in_sizes, int n_in,
                              void* d_out, int out_size, void* d_ws, size_t ws_size,
                              hipStream_t stream) {
    (void)in_sizes; (void)n_in; (void)out_size; (void)d_ws; (void)ws_size;
    const float* x = (const float*)d_in[0];
    const float* w = (const float*)d_in[1];
    float* out     = (float*)d_out;
    dim3 grid(8 * 16 * 4);   // b x cchunk x hseg
    dim3 block(256);         // 8 wave32 waves
    LocalRmsNorm_kernel<<<grid, block, 0, stream>>>(x, w, out);
}